// DetectionLayer_34729105555478
// MI455X (gfx1250) — compile-verified
//
#include <hip/hip_runtime.h>
#include <stdint.h>

// ---- problem constants (fixed by the reference) ----
#define GG        52
#define G2        (GG * GG)        // 2704
#define NA        3                // anchors
#define ATTRS     85               // 5 + 80
#define NBATCH    32
#define STRIDE_PX 8.0f             // 416 / 52

// ---- tiling ----
#define TILE_S    104              // 2 full gj rows; G2 / TILE_S == 26 exactly
#define NTILES    (G2 / TILE_S)    // 26
#define PITCH     108              // LDS row pitch (floats); 108*4 = 432 = 27*16 -> 16B aligned rows
#define NCHUNK    (TILE_S / 4)     // 26 x b128 transfers per channel row
#define NTHREADS  256

#if defined(__has_builtin)
#  if __has_builtin(__builtin_amdgcn_s_wait_asynccnt)
#    define WAIT_ASYNC0() __builtin_amdgcn_s_wait_asynccnt(0)
#  else
#    define WAIT_ASYNC0() asm volatile("s_wait_asynccnt 0" ::: "memory")
#  endif
#else
#  define WAIT_ASYNC0() asm volatile("s_wait_asynccnt 0" ::: "memory")
#endif

__global__ __launch_bounds__(NTHREADS) void yolo_decode_kernel(
    const float* __restrict__ pred,     // [B, A*85, 52, 52]
    const float* __restrict__ anchors,  // [A, 2]
    float* __restrict__ out)            // [B, G*G*A, 85]
{
    __shared__ __align__(16) float tile[ATTRS * PITCH];

    const int tid     = threadIdx.x;
    const int bi      = blockIdx.x;
    const int tileIdx = bi % NTILES;
    const int pair    = bi / NTILES;     // b * NA + a
    const int a       = pair % NA;
    const int b       = pair / NA;
    const int s0      = tileIdx * TILE_S;

    // base of this (b, a) slab: 85 channel planes of G2 floats each
    const float* gbase = pred + (size_t)pair * (size_t)(ATTRS * G2);
    // LDS byte address of tile[] (low 32 bits of generic pointer = DS offset)
    const uint32_t lbase = (uint32_t)(uintptr_t)(&tile[0]);

    // ---------------- Phase 1: async global -> LDS (coalesced reads) ----------------
    // 85 channels x 26 chunks of 16B; address = gbase + (c*G2 + s0 + 4q)*4 bytes
    for (int e = tid; e < ATTRS * NCHUNK; e += NTHREADS) {
        const int c = e / NCHUNK;
        const int q = e - c * NCHUNK;
        const uint32_t goff  = (uint32_t)((c * G2 + s0 + q * 4) * 4);
        const uint32_t laddr = lbase + (uint32_t)((c * PITCH + q * 4) * 4);
        asm volatile("global_load_async_to_lds_b128 %0, %1, %2"
                     :
                     : "v"(laddr), "v"(goff), "s"(gbase)
                     : "memory");
    }
    WAIT_ASYNC0();
    __syncthreads();

    // anchors in pixel units: bw = exp(t) * (anchors/stride) * stride = exp(t) * anchors
    const float aw = anchors[a * 2 + 0];
    const float ah = anchors[a * 2 + 1];

    // ---------------- Phase 2: transform + coalesced attribute-major stores ----------
    // out index = ((b*G2 + s)*NA + a)*ATTRS + c ; per spatial row 85 contiguous floats
    float* obase = out + (((size_t)b * G2 + s0) * NA + (size_t)a) * ATTRS;

    for (int o = tid; o < TILE_S * ATTRS; o += NTHREADS) {
        const int sl = o / ATTRS;          // local spatial position
        const int c  = o - sl * ATTRS;     // attribute
        const float x = tile[c * PITCH + sl];
        const int s = s0 + sl;

        float r;
        if (c >= 4) {
            // objectness + class scores: sigmoid
            r = 1.0f / (1.0f + __expf(-x));
        } else if (c == 0) {
            const float gi = (float)(s % GG);          // x offset varies along gi
            r = (1.0f / (1.0f + __expf(-x)) + gi) * STRIDE_PX;
        } else if (c == 1) {
            const float gj = (float)(s / GG);          // y offset varies along gj
            r = (1.0f / (1.0f + __expf(-x)) + gj) * STRIDE_PX;
        } else if (c == 2) {
            r = __expf(x) * aw;                        // width in pixels
        } else {
            r = __expf(x) * ah;                        // height in pixels
        }

        obase[(size_t)sl * (NA * ATTRS) + c] = r;
    }
}

extern "C" void kernel_launch(void* const* d_in, const int* in_sizes, int n_in,
                              void* d_out, int out_size, void* d_ws, size_t ws_size,
                              hipStream_t stream) {
    (void)in_sizes; (void)n_in; (void)out_size; (void)d_ws; (void)ws_size;
    const float* pred    = (const float*)d_in[0];
    const float* anchors = (const float*)d_in[1];
    float* out           = (float*)d_out;

    const dim3 grid(NBATCH * NA * NTILES);   // 32*3*26 = 2496 blocks
    const dim3 block(NTHREADS);
    yolo_decode_kernel<<<grid, block, 0, stream>>>(pred, anchors, out);
}